// CustomLoss_71244917506189
// MI455X (gfx1250) — compile-verified
//
#include <hip/hip_runtime.h>
#include <math.h>

typedef __attribute__((ext_vector_type(2))) float v2f;
typedef __attribute__((ext_vector_type(8))) float v8f;

#define B_ROWS 8192
#define D_COLS 4096
#define NT     (D_COLS / 16)        // 256 column tiles of 16
#define NC     32                   // batch chunks
#define CHUNK_ROWS (B_ROWS / NC)    // 256 rows per chunk

#define NEG_HALF_LOG2PI (-0.91893853320467274f)

// ---- workspace layout (floats) ----
#define WS_STATS 0                        // [0]=cnt0 [1]=cnt1 [2]=ldsum0 [3]=ldsum1
#define WS_MU    4                        // 2*D
#define WS_LSD   (WS_MU  + 2 * D_COLS)    // 2*D
#define WS_IV    (WS_LSD + 2 * D_COLS)    // 2*D : exp(-2*lsd)
#define WS_LP    (WS_IV  + 2 * D_COLS)    // B
#define WS_PART  (WS_LP  + B_ROWS)        // NC*NT*64 partial column sums

// ---------------------------------------------------------------------------
// k0: class counts + masked logdet sums (single block, fixed-order tree)
// ---------------------------------------------------------------------------
__global__ void __launch_bounds__(256)
k0_stats(const int* __restrict__ target, const float* __restrict__ logdet,
         float* __restrict__ stats) {
    __shared__ float s[256][4];
    const int tid = threadIdx.x;
    float c0 = 0.f, c1 = 0.f, l0 = 0.f, l1 = 0.f;
    for (int i = tid; i < B_ROWS; i += 256) {
        const int   t  = target[i];
        const float ld = logdet[i];
        if (t == 0) { c0 += 1.f; l0 += ld; }
        else        { c1 += 1.f; l1 += ld; }
    }
    s[tid][0] = c0; s[tid][1] = c1; s[tid][2] = l0; s[tid][3] = l1;
    __syncthreads();
    for (int off = 128; off > 0; off >>= 1) {
        if (tid < off)
            for (int j = 0; j < 4; ++j) s[tid][j] += s[tid + off][j];
        __syncthreads();
    }
    if (tid == 0) {
        stats[0] = s[0][0]; stats[1] = s[0][1];
        stats[2] = s[0][2]; stats[3] = s[0][3];
    }
}

// ---------------------------------------------------------------------------
// k1: masked column sums via V_WMMA_F32_16X16X4_F32.
//   A (16x4) = one-hot class masks: row0 = [t==0], row1 = [t==1], rest zero.
//   B (4x16) = 4-row x 16-col tile of mean / log_sd.
//   C rows 0/1 accumulate per-class column sums.
// One wave per block => EXEC all ones for WMMA.
// ---------------------------------------------------------------------------
__global__ void __launch_bounds__(32)
k1_colsum_wmma(const float* __restrict__ mean, const float* __restrict__ lsd,
               const int* __restrict__ target, float* __restrict__ partial) {
    const int tile  = blockIdx.x;          // 0..NT-1
    const int chunk = blockIdx.y;          // 0..NC-1
    const int lane  = threadIdx.x;         // 0..31
    const int m     = lane & 15;           // A row index / B column-in-tile
    const int k0    = (lane >> 4) << 1;    // K pair this half-wave holds: 0 or 2
    const int col   = tile * 16 + m;       // global column this lane holds for B

    v8f cm = {};   // accumulates class sums of `mean`
    v8f cl = {};   // accumulates class sums of `log_sd`
    int rb = chunk * CHUNK_ROWS;
    const size_t D = (size_t)D_COLS;

    for (int it = 0; it < CHUNK_ROWS / 4; ++it, rb += 4) {
        // A layout (32-bit 16x4): lanes 0-15 hold K=0(x),K=1(y); lanes 16-31 K=2(x),K=3(y)
        v2f a; a.x = 0.f; a.y = 0.f;
        if (m < 2) {
            a.x = (target[rb + k0]     == m) ? 1.f : 0.f;
            a.y = (target[rb + k0 + 1] == m) ? 1.f : 0.f;
        }
        // B layout (4x16): VGPR0 = row k0 (this half-wave), VGPR1 = row k0+1; N = lane&15
        const float* pm = mean + (size_t)(rb + k0) * D + col;
        const float* pl = lsd  + (size_t)(rb + k0) * D + col;
        v2f bm; bm.x = pm[0]; bm.y = pm[D];
        v2f bl; bl.x = pl[0]; bl.y = pl[D];

        cm = __builtin_amdgcn_wmma_f32_16x16x4_f32(false, a, false, bm, (short)0, cm,
                                                   false, false);
        cl = __builtin_amdgcn_wmma_f32_16x16x4_f32(false, a, false, bl, (short)0, cl,
                                                   false, false);
    }

    // C/D layout: VGPR r, lanes 0-15 => row M=r, col N=lane. Rows 0/1 are our classes.
    if (lane < 16) {
        const size_t base = ((size_t)(chunk * NT + tile) * 4) * 16 + lane;
        partial[base +  0] = cm[0];   // q=0: mean, class 0
        partial[base + 16] = cm[1];   // q=1: mean, class 1
        partial[base + 32] = cl[0];   // q=2: log_sd, class 0
        partial[base + 48] = cl[1];   // q=3: log_sd, class 1
    }
}

// ---------------------------------------------------------------------------
// k2: reduce chunk partials (fixed order), divide by counts, write mu / lsd
//     to d_out, stash mu / lsd / exp(-2*lsd) in workspace for phase 2.
// ---------------------------------------------------------------------------
__global__ void __launch_bounds__(256)
k2_params(const float* __restrict__ partial, const float* __restrict__ stats,
          float* __restrict__ out, float* __restrict__ ws_mu,
          float* __restrict__ ws_lsd, float* __restrict__ ws_iv) {
    const int idx = blockIdx.x * blockDim.x + threadIdx.x;   // 0 .. 4*D-1
    if (idx >= 4 * D_COLS) return;
    const int q    = idx / D_COLS;          // 0:mu_c0 1:mu_c1 2:lsd_c0 3:lsd_c1
    const int col  = idx - q * D_COLS;
    const int tile = col >> 4;
    const int n    = col & 15;

    const size_t base   = ((size_t)tile * 4 + q) * 16 + n;
    const size_t stride = (size_t)NT * 64;
    float s = 0.f;
    for (int c = 0; c < NC; ++c) s += partial[base + (size_t)c * stride];

    const int   cls = q & 1;
    const float v   = s / stats[cls];
    if (q < 2) {
        out[1 + q * D_COLS + col] = v;                 // mus_per_class
        ws_mu[q * D_COLS + col]   = v;
    } else {
        const int j = q - 2;
        out[1 + 2 * D_COLS + j * D_COLS + col] = v;    // log_sds_per_class
        ws_lsd[j * D_COLS + col] = v;
        ws_iv[j * D_COLS + col]  = expf(-2.f * v);
    }
}

// ---------------------------------------------------------------------------
// k3: per-row gaussian log-prob summed over features; one wave per row,
//     float4 streaming loads of z, wave shuffle reduction.
// ---------------------------------------------------------------------------
__global__ void __launch_bounds__(256)
k3_rowlp(const float* __restrict__ z, const int* __restrict__ target,
         const float* __restrict__ ws_mu, const float* __restrict__ ws_lsd,
         const float* __restrict__ ws_iv, float* __restrict__ lp) {
    const int wave = threadIdx.x >> 5;
    const int lane = threadIdx.x & 31;
    const int row  = blockIdx.x * 8 + wave;
    const int t    = target[row];

    const float* zr = z      + (size_t)row * D_COLS;
    const float* mu = ws_mu  + (size_t)t   * D_COLS;
    const float* ls = ws_lsd + (size_t)t   * D_COLS;
    const float* iv = ws_iv  + (size_t)t   * D_COLS;

    float acc = 0.f;
    for (int col = lane * 4; col < D_COLS; col += 128) {
        const float4 zv = *(const float4*)(zr + col);
        const float4 mv = *(const float4*)(mu + col);
        const float4 lv = *(const float4*)(ls + col);
        const float4 vv = *(const float4*)(iv + col);
        const float d0 = zv.x - mv.x, d1 = zv.y - mv.y;
        const float d2 = zv.z - mv.z, d3 = zv.w - mv.w;
        acc += (NEG_HALF_LOG2PI - lv.x - 0.5f * d0 * d0 * vv.x);
        acc += (NEG_HALF_LOG2PI - lv.y - 0.5f * d1 * d1 * vv.y);
        acc += (NEG_HALF_LOG2PI - lv.z - 0.5f * d2 * d2 * vv.z);
        acc += (NEG_HALF_LOG2PI - lv.w - 0.5f * d3 * d3 * vv.w);
    }
    for (int off = 16; off > 0; off >>= 1) acc += __shfl_xor(acc, off, 32);
    if (lane == 0) lp[row] = acc;
}

// ---------------------------------------------------------------------------
// k4: masked means of lp per class + prior scalar (single block, fixed order)
// ---------------------------------------------------------------------------
__global__ void __launch_bounds__(256)
k4_final(const float* __restrict__ lp, const int* __restrict__ target,
         const float* __restrict__ stats, float* __restrict__ out) {
    __shared__ float s0[256], s1[256];
    const int tid = threadIdx.x;
    float a0 = 0.f, a1 = 0.f;
    for (int i = tid; i < B_ROWS; i += 256) {
        const float v = lp[i];
        if (target[i] == 0) a0 += v; else a1 += v;
    }
    s0[tid] = a0; s1[tid] = a1;
    __syncthreads();
    for (int off = 128; off > 0; off >>= 1) {
        if (tid < off) { s0[tid] += s0[tid + off]; s1[tid] += s1[tid + off]; }
        __syncthreads();
    }
    if (tid == 0) {
        const float cnt0 = stats[0], cnt1 = stats[1];
        const float lp0 = s0[0] / cnt0, lp1 = s1[0] / cnt1;
        const float ldm0 = stats[2] / cnt0, ldm1 = stats[3] / cnt1;
        out[0]              = 0.5f * ((lp0 + ldm0) + (lp1 + ldm1));  // prior_logprob
        out[1 + 4 * D_COLS] = lp0;                                   // log_p_total[0]
        out[2 + 4 * D_COLS] = lp1;                                   // log_p_total[1]
    }
}

// ---------------------------------------------------------------------------
extern "C" void kernel_launch(void* const* d_in, const int* in_sizes, int n_in,
                              void* d_out, int out_size, void* d_ws, size_t ws_size,
                              hipStream_t stream) {
    const float* z      = (const float*)d_in[0];
    const float* mean   = (const float*)d_in[1];
    const float* log_sd = (const float*)d_in[2];
    const int*   target = (const int*)  d_in[3];
    const float* logdet = (const float*)d_in[4];
    float* out = (float*)d_out;
    float* ws  = (float*)d_ws;

    float* stats  = ws + WS_STATS;
    float* ws_mu  = ws + WS_MU;
    float* ws_lsd = ws + WS_LSD;
    float* ws_iv  = ws + WS_IV;
    float* lp     = ws + WS_LP;
    float* part   = ws + WS_PART;

    k0_stats<<<1, 256, 0, stream>>>(target, logdet, stats);

    k1_colsum_wmma<<<dim3(NT, NC), 32, 0, stream>>>(mean, log_sd, target, part);

    k2_params<<<(4 * D_COLS + 255) / 256, 256, 0, stream>>>(part, stats, out,
                                                            ws_mu, ws_lsd, ws_iv);

    k3_rowlp<<<B_ROWS / 8, 256, 0, stream>>>(z, target, ws_mu, ws_lsd, ws_iv, lp);

    k4_final<<<1, 256, 0, stream>>>(lp, target, stats, out);
}